// TransformerBlock_34711925686362
// MI455X (gfx1250) — compile-verified
//
#include <hip/hip_runtime.h>
#include <hip/hip_bf16.h>
#include <math.h>

#define N_   50000
#define E_   1600000
#define IN_  128
#define HID_ 128
#define H_   8
#define C_   16
#define ED_  32

typedef float v2f __attribute__((ext_vector_type(2)));
typedef float v8f __attribute__((ext_vector_type(8)));

__device__ __forceinline__ v8f wmma_f32_k4(v2f a, v2f b, v8f c) {
    return __builtin_amdgcn_wmma_f32_16x16x4_f32(false, a, false, b, (short)0, c, false, false);
}

__device__ __forceinline__ void atomicMaxFloat(float* addr, float val) {
    if (val >= 0.0f) atomicMax((int*)addr, __float_as_int(val));
    else             atomicMin((unsigned int*)addr, __float_as_uint(val));
}

__device__ __forceinline__ float sigmoidf(float x) {
    return 1.0f / (1.0f + expf(-x));
}

// ---------------------------------------------------------------- init
__global__ void init_kernel(float* out, float* m, float* s, float* hsum) {
    size_t i = (size_t)blockIdx.x * blockDim.x + threadIdx.x;
    if (i < (size_t)N_ * HID_) out[i] = 0.0f;
    if (i < (size_t)N_ * H_) { m[i] = -3.0e38f; s[i] = 0.0f; }
    if (i < HID_) hsum[i] = 0.0f;
}

// ------------------------------------------------- fused q/k/v/xr GEMM (WMMA f32)
// grid: N/16 blocks x 256 threads. Each wave owns ONE weight matrix
// (mat = wave>>1) and 4 column tiles of it ((wave&1)*64 .. +64).
__global__ __launch_bounds__(256)
void node_qkv_kernel(const float* __restrict__ x,
                     const float* __restrict__ Wq, const float* __restrict__ bq,
                     const float* __restrict__ Wk, const float* __restrict__ bk,
                     const float* __restrict__ Wv, const float* __restrict__ bv,
                     const float* __restrict__ Wsk, const float* __restrict__ bsk,
                     float* __restrict__ q, float* __restrict__ k,
                     float* __restrict__ v, float* __restrict__ xr) {
    __shared__ float xs[16 * 132];
    const int row0 = blockIdx.x * 16;
    for (int i = threadIdx.x; i < 16 * 128; i += 256) {
        int r = i >> 7, c = i & 127;
        xs[r * 132 + c] = x[(size_t)(row0 + r) * IN_ + c];
    }
    __syncthreads();

    const int wave = __builtin_amdgcn_readfirstlane(threadIdx.x >> 5);
    const int lane = threadIdx.x & 31;
    const int half = lane >> 4;   // 0: K even pair, 1: K odd pair
    const int l16  = lane & 15;

    // wave-uniform (scalar) matrix selection, hoisted out of all loops
    const int mat   = wave >> 1;          // 0..3
    const int cbase = (wave & 1) * 64;    // 0 or 64
    const float* __restrict__ Wm = (mat == 0) ? Wq : (mat == 1) ? Wk : (mat == 2) ? Wv : Wsk;
    const float* __restrict__ Bm = (mat == 0) ? bq : (mat == 1) ? bk : (mat == 2) ? bv : bsk;
    float* __restrict__       Om = (mat == 0) ? q  : (mat == 1) ? k  : (mat == 2) ? v  : xr;

    v8f acc[4];
    acc[0] = {}; acc[1] = {}; acc[2] = {}; acc[3] = {};

    for (int kk = 0; kk < 128; kk += 4) {
        v2f a;
        a[0] = xs[l16 * 132 + kk + half * 2];
        a[1] = xs[l16 * 132 + kk + half * 2 + 1];
#pragma unroll
        for (int t = 0; t < 4; ++t) {
            const int c0 = cbase + t * 16;
            v2f b;
            b[0] = Wm[(kk + half * 2) * HID_ + c0 + l16];
            b[1] = Wm[(kk + half * 2 + 1) * HID_ + c0 + l16];
            acc[t] = wmma_f32_k4(a, b, acc[t]);
        }
    }

#pragma unroll
    for (int t = 0; t < 4; ++t) {
        const int c0 = cbase + t * 16;
        const float bias = Bm[c0 + l16];
#pragma unroll
        for (int r = 0; r < 8; ++r) {
            const int row = row0 + r + half * 8;
            Om[(size_t)row * HID_ + c0 + l16] = acc[t][r] + bias;
        }
    }
}

// ------------------------------------------------- edge pass A: alpha + segment max
// one wave per edge; lane j owns hid channels 4j..4j+3 (head = j>>2)
__global__ __launch_bounds__(256)
void edge_alpha_kernel(const int* __restrict__ ei, const float* __restrict__ ef,
                       const float* __restrict__ We, const float* __restrict__ be,
                       const float* __restrict__ q, const float* __restrict__ k,
                       float* __restrict__ alpha, float* __restrict__ m) {
    __shared__ float sWe[ED_ * HID_];
    __shared__ float sbe[HID_];
    for (int i = threadIdx.x; i < ED_ * HID_; i += 256) sWe[i] = We[i];
    if (threadIdx.x < HID_) sbe[threadIdx.x] = be[threadIdx.x];
    __syncthreads();

    const int wave = __builtin_amdgcn_readfirstlane(threadIdx.x >> 5);
    const int lane = threadIdx.x & 31;
    const int e    = blockIdx.x * 8 + wave;                 // scalar
    const int src  = __builtin_amdgcn_readfirstlane(ei[e]); // scalar
    const int dst  = __builtin_amdgcn_readfirstlane(ei[E_ + e]);

    const float efj = ef[(size_t)e * ED_ + lane];
    float4 ev = *(const float4*)&sbe[4 * lane];
#pragma unroll
    for (int d = 0; d < ED_; ++d) {
        const float efd = __shfl(efj, d, 32);
        const float4 w4 = *(const float4*)&sWe[d * HID_ + 4 * lane];
        ev.x += efd * w4.x; ev.y += efd * w4.y; ev.z += efd * w4.z; ev.w += efd * w4.w;
    }

    const float4 q4 = *(const float4*)(q + (size_t)dst * HID_ + 4 * lane);
    const float4 k4 = *(const float4*)(k + (size_t)src * HID_ + 4 * lane);
    float p = q4.x * (k4.x + ev.x) + q4.y * (k4.y + ev.y)
            + q4.z * (k4.z + ev.z) + q4.w * (k4.w + ev.w);
    p += __shfl_xor(p, 1);
    p += __shfl_xor(p, 2);
    p *= 0.25f;                          // 1/sqrt(C), C=16

    if ((lane & 3) == 0) {
        const int head = lane >> 2;
        alpha[(size_t)e * H_ + head] = p;
        atomicMaxFloat(&m[(size_t)dst * H_ + head], p);
    }
}

// ------------------------------------------------- exp + segment sum
__global__ void edge_exp_kernel(const int* __restrict__ ei, float* __restrict__ alpha,
                                const float* __restrict__ m, float* __restrict__ s) {
    const size_t i = (size_t)blockIdx.x * blockDim.x + threadIdx.x;
    if (i >= (size_t)E_ * H_) return;
    const int e = (int)(i >> 3);
    const int head = (int)(i & 7);
    const int dst = ei[E_ + e];
    const float ex = expf(alpha[i] - m[(size_t)dst * H_ + head]);
    alpha[i] = ex;
    atomicAdd(&s[(size_t)dst * H_ + head], ex);
}

// ------------------------------------------------- edge pass B: weighted scatter-add
__global__ __launch_bounds__(256)
void edge_msg_kernel(const int* __restrict__ ei, const float* __restrict__ ef,
                     const float* __restrict__ We, const float* __restrict__ be,
                     const float* __restrict__ v, const float* __restrict__ alpha,
                     const float* __restrict__ s, float* __restrict__ out) {
    __shared__ float sWe[ED_ * HID_];
    __shared__ float sbe[HID_];
    for (int i = threadIdx.x; i < ED_ * HID_; i += 256) sWe[i] = We[i];
    if (threadIdx.x < HID_) sbe[threadIdx.x] = be[threadIdx.x];
    __syncthreads();

    const int wave = __builtin_amdgcn_readfirstlane(threadIdx.x >> 5);
    const int lane = threadIdx.x & 31;
    const int e    = blockIdx.x * 8 + wave;
    const int src  = __builtin_amdgcn_readfirstlane(ei[e]);
    const int dst  = __builtin_amdgcn_readfirstlane(ei[E_ + e]);

    const float efj = ef[(size_t)e * ED_ + lane];
    float4 ev = *(const float4*)&sbe[4 * lane];
#pragma unroll
    for (int d = 0; d < ED_; ++d) {
        const float efd = __shfl(efj, d, 32);
        const float4 w4 = *(const float4*)&sWe[d * HID_ + 4 * lane];
        ev.x += efd * w4.x; ev.y += efd * w4.y; ev.z += efd * w4.z; ev.w += efd * w4.w;
    }

    const int head = lane >> 2;
    const float ex = alpha[(size_t)e * H_ + head];
    const float an = ex / (s[(size_t)dst * H_ + head] + 1e-16f);

    const float4 v4 = *(const float4*)(v + (size_t)src * HID_ + 4 * lane);
    float* ob = out + (size_t)dst * HID_ + 4 * lane;
    atomicAdd(ob + 0, (v4.x + ev.x) * an);
    atomicAdd(ob + 1, (v4.y + ev.y) * an);
    atomicAdd(ob + 2, (v4.z + ev.z) * an);
    atomicAdd(ob + 3, (v4.w + ev.w) * an);
}

// ------------------------------------------------- beta skip + LN1 + ReLU -> h
__global__ __launch_bounds__(256)
void node_combine_kernel(const float* __restrict__ x, const float* __restrict__ out,
                         const float* __restrict__ xr, const float* __restrict__ Wbeta,
                         const float* __restrict__ g1, const float* __restrict__ b1,
                         float* __restrict__ h) {
    const int wave = __builtin_amdgcn_readfirstlane(threadIdx.x >> 5);
    const int lane = threadIdx.x & 31;
    const int n = blockIdx.x * 8 + wave;
    const int j4 = 4 * lane;

    const float4 o4 = *(const float4*)(out + (size_t)n * HID_ + j4);
    const float4 r4 = *(const float4*)(xr  + (size_t)n * HID_ + j4);
    const float4 x4 = *(const float4*)(x   + (size_t)n * IN_  + j4);

    float p = 0.0f;
    p += o4.x * Wbeta[j4 + 0] + r4.x * Wbeta[128 + j4 + 0] + (o4.x - r4.x) * Wbeta[256 + j4 + 0];
    p += o4.y * Wbeta[j4 + 1] + r4.y * Wbeta[128 + j4 + 1] + (o4.y - r4.y) * Wbeta[256 + j4 + 1];
    p += o4.z * Wbeta[j4 + 2] + r4.z * Wbeta[128 + j4 + 2] + (o4.z - r4.z) * Wbeta[256 + j4 + 2];
    p += o4.w * Wbeta[j4 + 3] + r4.w * Wbeta[128 + j4 + 3] + (o4.w - r4.w) * Wbeta[256 + j4 + 3];
#pragma unroll
    for (int off = 16; off > 0; off >>= 1) p += __shfl_xor(p, off);
    const float beta = sigmoidf(p);

    float4 t;
    t.x = beta * r4.x + (1.0f - beta) * o4.x + x4.x;
    t.y = beta * r4.y + (1.0f - beta) * o4.y + x4.y;
    t.z = beta * r4.z + (1.0f - beta) * o4.z + x4.z;
    t.w = beta * r4.w + (1.0f - beta) * o4.w + x4.w;

    float sum = t.x + t.y + t.z + t.w;
#pragma unroll
    for (int off = 16; off > 0; off >>= 1) sum += __shfl_xor(sum, off);
    const float mean = sum * (1.0f / 128.0f);

    float4 d = { t.x - mean, t.y - mean, t.z - mean, t.w - mean };
    float sq = d.x * d.x + d.y * d.y + d.z * d.z + d.w * d.w;
#pragma unroll
    for (int off = 16; off > 0; off >>= 1) sq += __shfl_xor(sq, off);
    const float inv = rsqrtf(sq * (1.0f / 128.0f) + 1e-5f);

    float4 hv;
    hv.x = fmaxf(d.x * inv * g1[j4 + 0] + b1[j4 + 0], 0.0f);
    hv.y = fmaxf(d.y * inv * g1[j4 + 1] + b1[j4 + 1], 0.0f);
    hv.z = fmaxf(d.z * inv * g1[j4 + 2] + b1[j4 + 2], 0.0f);
    hv.w = fmaxf(d.w * inv * g1[j4 + 3] + b1[j4 + 3], 0.0f);
    *(float4*)(h + (size_t)n * HID_ + j4) = hv;
}

// ------------------------------------------------- column sum of h
__global__ void colsum_kernel(const float* __restrict__ h, float* __restrict__ hsum) {
    const int col = threadIdx.x;             // 128 threads
    const int row0 = blockIdx.x * 256;
    float acc = 0.0f;
    for (int r = 0; r < 256; ++r) {
        const int row = row0 + r;
        if (row < N_) acc += h[(size_t)row * HID_ + col];
    }
    atomicAdd(&hsum[col], acc);
}

// ------------------------------------------------- global context (1 block, 128 thr)
__global__ void global_ctx_kernel(const float* __restrict__ hsum,
                                  const float* __restrict__ Wr, const float* __restrict__ br,
                                  const float* __restrict__ Ww, const float* __restrict__ bw,
                                  const float* __restrict__ Wg, const float* __restrict__ bg,
                                  float* __restrict__ gbc_out, float* __restrict__ gconst_out) {
    __shared__ float sg[HID_];
    __shared__ float sgbc[HID_];
    const int j = threadIdx.x;
    float acc = br[j];
    for (int d = 0; d < HID_; ++d) acc += (hsum[d] * (1.0f / (float)N_)) * Wr[d * HID_ + j];
    sg[j] = fmaxf(acc, 0.0f);
    __syncthreads();
    float acc2 = bw[j];
    for (int d = 0; d < HID_; ++d) acc2 += sg[d] * Ww[d * HID_ + j];
    sgbc[j] = acc2;
    gbc_out[j] = acc2;
    __syncthreads();
    float acc3 = bg[j];
    for (int d = 0; d < HID_; ++d) acc3 += sgbc[d] * Wg[(HID_ + d) * HID_ + j];
    gconst_out[j] = acc3;
}

// ------------------------------------------------- gate GEMM (WMMA) + LN2 -> d_out
__global__ __launch_bounds__(256)
void final_kernel(const float* __restrict__ h, const float* __restrict__ Wg,
                  const float* __restrict__ gconst, const float* __restrict__ gbc,
                  const float* __restrict__ g2, const float* __restrict__ b2,
                  float* __restrict__ y) {
    __shared__ float hs[16 * 132];
    __shared__ float lg[16 * 132];
    const int row0 = blockIdx.x * 16;
    for (int i = threadIdx.x; i < 16 * 128; i += 256) {
        int r = i >> 7, c = i & 127;
        hs[r * 132 + c] = h[(size_t)(row0 + r) * HID_ + c];
    }
    __syncthreads();

    const int wave = __builtin_amdgcn_readfirstlane(threadIdx.x >> 5);
    const int lane = threadIdx.x & 31;
    const int half = lane >> 4;
    const int l16  = lane & 15;
    const int c0   = wave * 16;

    v8f acc = {};
    for (int kk = 0; kk < 128; kk += 4) {
        v2f a;
        a[0] = hs[l16 * 132 + kk + half * 2];
        a[1] = hs[l16 * 132 + kk + half * 2 + 1];
        v2f b;
        b[0] = Wg[(kk + half * 2) * HID_ + c0 + l16];
        b[1] = Wg[(kk + half * 2 + 1) * HID_ + c0 + l16];
        acc = wmma_f32_k4(a, b, acc);
    }
#pragma unroll
    for (int r = 0; r < 8; ++r)
        lg[(r + half * 8) * 132 + c0 + l16] = acc[r];
    __syncthreads();

    // LN over each of the 16 rows; wave w handles rows 2w, 2w+1
    for (int rr = wave * 2; rr < wave * 2 + 2; ++rr) {
        const int j4 = 4 * lane;
        float t[4];
#pragma unroll
        for (int c = 0; c < 4; ++c) {
            const float hv = hs[rr * 132 + j4 + c];
            const float gate = sigmoidf(lg[rr * 132 + j4 + c] + gconst[j4 + c]);
            t[c] = hv + gate * gbc[j4 + c];
        }
        float sum = t[0] + t[1] + t[2] + t[3];
#pragma unroll
        for (int off = 16; off > 0; off >>= 1) sum += __shfl_xor(sum, off);
        const float mean = sum * (1.0f / 128.0f);
        float sq = 0.0f;
#pragma unroll
        for (int c = 0; c < 4; ++c) { t[c] -= mean; sq += t[c] * t[c]; }
#pragma unroll
        for (int off = 16; off > 0; off >>= 1) sq += __shfl_xor(sq, off);
        const float inv = rsqrtf(sq * (1.0f / 128.0f) + 1e-5f);
        float4 o;
        o.x = t[0] * inv * g2[j4 + 0] + b2[j4 + 0];
        o.y = t[1] * inv * g2[j4 + 1] + b2[j4 + 1];
        o.z = t[2] * inv * g2[j4 + 2] + b2[j4 + 2];
        o.w = t[3] * inv * g2[j4 + 3] + b2[j4 + 3];
        *(float4*)(y + (size_t)(row0 + rr) * HID_ + j4) = o;
    }
}

// ----------------------------------------------------------------------------
extern "C" void kernel_launch(void* const* d_in, const int* in_sizes, int n_in,
                              void* d_out, int out_size, void* d_ws, size_t ws_size,
                              hipStream_t stream) {
    const float* x     = (const float*)d_in[0];
    const int*   ei    = (const int*)d_in[1];
    const float* ef    = (const float*)d_in[2];
    const float* Wq    = (const float*)d_in[3];
    const float* bq    = (const float*)d_in[4];
    const float* Wk    = (const float*)d_in[5];
    const float* bk    = (const float*)d_in[6];
    const float* Wv    = (const float*)d_in[7];
    const float* bv    = (const float*)d_in[8];
    const float* We    = (const float*)d_in[9];
    const float* be    = (const float*)d_in[10];
    const float* Wsk   = (const float*)d_in[11];
    const float* bsk   = (const float*)d_in[12];
    const float* Wbeta = (const float*)d_in[13];
    const float* ln1g  = (const float*)d_in[14];
    const float* ln1b  = (const float*)d_in[15];
    const float* Wr    = (const float*)d_in[16];
    const float* br    = (const float*)d_in[17];
    const float* Ww    = (const float*)d_in[18];
    const float* bw    = (const float*)d_in[19];
    const float* Wg    = (const float*)d_in[20];
    const float* bg    = (const float*)d_in[21];
    const float* ln2g  = (const float*)d_in[22];
    const float* ln2b  = (const float*)d_in[23];

    float* ws = (float*)d_ws;
    const size_t NH = (size_t)N_ * HID_;
    float* q      = ws;
    float* k      = q + NH;
    float* v      = k + NH;
    float* xr     = v + NH;
    float* outb   = xr + NH;
    float* h      = outb + NH;
    float* alpha  = h + NH;                  // E*H
    float* m      = alpha + (size_t)E_ * H_; // N*H
    float* s      = m + (size_t)N_ * H_;     // N*H
    float* hsum   = s + (size_t)N_ * H_;     // HID
    float* gbc    = hsum + HID_;             // HID
    float* gconst = gbc + HID_;              // HID
    float* y      = (float*)d_out;

    // 1. init accumulators
    {
        const size_t total = NH;
        const int blocks = (int)((total + 255) / 256);
        init_kernel<<<blocks, 256, 0, stream>>>(outb, m, s, hsum);
    }
    // 2. q,k,v,xr GEMMs (WMMA f32)
    node_qkv_kernel<<<N_ / 16, 256, 0, stream>>>(x, Wq, bq, Wk, bk, Wv, bv, Wsk, bsk,
                                                 q, k, v, xr);
    // 3. edge attention logits + segment max
    edge_alpha_kernel<<<E_ / 8, 256, 0, stream>>>(ei, ef, We, be, q, k, alpha, m);
    // 4. exp + segment sum
    edge_exp_kernel<<<(int)(((size_t)E_ * H_ + 255) / 256), 256, 0, stream>>>(ei, alpha, m, s);
    // 5. weighted message scatter-add
    edge_msg_kernel<<<E_ / 8, 256, 0, stream>>>(ei, ef, We, be, v, alpha, s, outb);
    // 6. beta skip + LN1 + ReLU
    node_combine_kernel<<<N_ / 8, 256, 0, stream>>>(x, outb, xr, Wbeta, ln1g, ln1b, h);
    // 7. column mean (sum) of h
    colsum_kernel<<<(N_ + 255) / 256, 128, 0, stream>>>(h, hsum);
    // 8. global context vectors
    global_ctx_kernel<<<1, 128, 0, stream>>>(hsum, Wr, br, Ww, bw, Wg, bg, gbc, gconst);
    // 9. gate GEMM (WMMA) + LN2 -> output
    final_kernel<<<N_ / 16, 256, 0, stream>>>(h, Wg, gconst, gbc, ln2g, ln2b, y);
}